// NeuralTexture_64922725646779
// MI455X (gfx1250) — compile-verified
//
#include <hip/hip_runtime.h>

// ---------------------------------------------------------------------------
// NeuralTexture grid_sample (bilinear, border, align_corners) for MI455X.
//   x:    [B=4, H=768, W=768, 2] f32 grid in [-1,1]
//   data: [C=16, T=1024, T=1024] f32 texture, clamped to [CLAMP_LO, CLAMP_HI]
//   out:  [B, C, H, W] f32
// Strategy: (1) clamp+transpose texture to [T, T, C] in workspace (64 MB,
// L2-resident on the 192 MB L2); (2) one thread per pixel, 4 taps x 4
// float4 channel-chunks via global_load_b128, NT-stream the 151 MB output so
// it never evicts the texture from L2. Grid tile staged to LDS via the
// CDNA5 async global->LDS path (builtin confirmed present by round-1 probe).
// ---------------------------------------------------------------------------

#define CLAMP_LO (-123.68f)
#define CLAMP_HI (151.061f)

constexpr int C_  = 16;
constexpr int T_  = 1024;
constexpr int B_  = 4;
constexpr int H_  = 768;
constexpr int W_  = 768;
constexpr int HW_ = H_ * W_;
constexpr int TT_ = T_ * T_;
constexpr int NPIX_ = B_ * HW_;          // 2,359,296  (multiple of 256)

static_assert(NPIX_ % 256 == 0, "pixel count divisible by block");
static_assert(TT_ % 256 == 0, "texel count divisible by block");

#if defined(__has_builtin)
#if __has_builtin(__builtin_amdgcn_global_load_async_to_lds_b64) && \
    __has_builtin(__builtin_amdgcn_s_wait_asynccnt)
#define USE_ASYNC_LDS 1
#endif
#endif
#ifndef USE_ASYNC_LDS
#define USE_ASYNC_LDS 0
#endif

// Round-1 diagnostic: param 0 of the async builtin is
//   '__attribute__((__vector_size__(2 * sizeof(int)))) int __device__ *'
// i.e. a non-const AS(1) pointer to a v2i; build matching pointer types.
typedef int v2i __attribute__((vector_size(2 * sizeof(int))));
typedef __attribute__((address_space(1))) v2i* gas_v2i;
typedef __attribute__((address_space(3))) v2i* las_v2i;

// ---- Kernel 1: clamp + transpose [C,T,T] -> [T,T,C] (AoS, 64B per texel) ---
__global__ __launch_bounds__(256) void tex_transpose_clamp(
    const float* __restrict__ data, float4* __restrict__ ws4) {
  const int gid = blockIdx.x * 256 + threadIdx.x;   // texel index y*T + x
  float v[16];
#pragma unroll
  for (int c = 0; c < 16; ++c) {
    // read-once: non-temporal so the raw texture doesn't fight the
    // transposed copy for L2 residency
    float t = __builtin_nontemporal_load(data + (size_t)c * TT_ + gid);
    v[c] = fminf(fmaxf(t, CLAMP_LO), CLAMP_HI);
  }
#pragma unroll
  for (int j = 0; j < 4; ++j) {
    float4 p;
    p.x = v[4 * j + 0]; p.y = v[4 * j + 1];
    p.z = v[4 * j + 2]; p.w = v[4 * j + 3];
    ws4[(size_t)gid * 4 + j] = p;   // RT store: leave hot in L2
  }
}

// ---- Kernel 2: bilinear sample from [T,T,C] workspace ----------------------
__global__ __launch_bounds__(256) void grid_sample_tex16(
    const float2* __restrict__ grid, const float4* __restrict__ ws4,
    float* __restrict__ out) {
  const int tid = threadIdx.x;
  const int gid = blockIdx.x * 256 + tid;           // pixel index

#if USE_ASYNC_LDS
  __shared__ float2 gtile[256];
  {
    gas_v2i gsrc = (gas_v2i)(uintptr_t)(const void*)(grid + gid);
    las_v2i ldst = (las_v2i)(unsigned)(uintptr_t)(void*)&gtile[tid];
    __builtin_amdgcn_global_load_async_to_lds_b64(gsrc, ldst, 0, 0);
  }
#endif

  // index math overlaps the (async) grid fetch
  const int b  = gid / HW_;
  const int hw = gid - b * HW_;
  const int obase = b * (16 * HW_) + hw;

  float gx, gy;
#if USE_ASYNC_LDS
  __builtin_amdgcn_s_wait_asynccnt(0);
  { float2 g = gtile[tid]; gx = g.x; gy = g.y; }
#else
  {
    const float* gp = (const float*)(grid + gid);   // read-once: NT
    gx = __builtin_nontemporal_load(gp + 0);
    gy = __builtin_nontemporal_load(gp + 1);
  }
#endif

  // (g+1)*0.5*(T-1) = g*511.5 + 511.5, then clip to [0, T-1]
  float ix = fminf(fmaxf(fmaf(gx, 511.5f, 511.5f), 0.f), 1023.f);
  float iy = fminf(fmaxf(fmaf(gy, 511.5f, 511.5f), 0.f), 1023.f);
  const float fx0 = floorf(ix), fy0 = floorf(iy);
  const float wx = ix - fx0,    wy = iy - fy0;
  const int x0 = (int)fx0, y0 = (int)fy0;
  const int x1 = min(x0 + 1, 1023), y1 = min(y0 + 1, 1023);
  const float wx1 = 1.f - wx, wy1 = 1.f - wy;
  const float w00 = wx1 * wy1, w01 = wx * wy1;
  const float w10 = wx1 * wy,  w11 = wx * wy;

  const float4* p00 = ws4 + (((y0 << 10) + x0) << 2);
  const float4* p01 = ws4 + (((y0 << 10) + x1) << 2);
  const float4* p10 = ws4 + (((y1 << 10) + x0) << 2);
  const float4* p11 = ws4 + (((y1 << 10) + x1) << 2);

#pragma unroll
  for (int j = 0; j < 4; ++j) {
    const float4 a = p00[j], bq = p01[j], cq = p10[j], dq = p11[j];
    float r0 = fmaf(w11, dq.x, fmaf(w10, cq.x, fmaf(w01, bq.x, w00 * a.x)));
    float r1 = fmaf(w11, dq.y, fmaf(w10, cq.y, fmaf(w01, bq.y, w00 * a.y)));
    float r2 = fmaf(w11, dq.z, fmaf(w10, cq.z, fmaf(w01, bq.z, w00 * a.z)));
    float r3 = fmaf(w11, dq.w, fmaf(w10, cq.w, fmaf(w01, bq.w, w00 * a.w)));
    // stream the 151 MB output: never pollute L2 (keeps texture resident)
    __builtin_nontemporal_store(r0, out + obase + (4 * j + 0) * HW_);
    __builtin_nontemporal_store(r1, out + obase + (4 * j + 1) * HW_);
    __builtin_nontemporal_store(r2, out + obase + (4 * j + 2) * HW_);
    __builtin_nontemporal_store(r3, out + obase + (4 * j + 3) * HW_);
  }
}

// ---- Fallback: direct sample from [C,T,T] (if ws too small) ----------------
__global__ __launch_bounds__(256) void grid_sample_direct(
    const float2* __restrict__ grid, const float* __restrict__ data,
    float* __restrict__ out) {
  const int gid = blockIdx.x * 256 + threadIdx.x;
  const int b  = gid / HW_;
  const int hw = gid - b * HW_;
  const float2 g = grid[gid];
  float ix = fminf(fmaxf(fmaf(g.x, 511.5f, 511.5f), 0.f), 1023.f);
  float iy = fminf(fmaxf(fmaf(g.y, 511.5f, 511.5f), 0.f), 1023.f);
  const float fx0 = floorf(ix), fy0 = floorf(iy);
  const float wx = ix - fx0, wy = iy - fy0;
  const int x0 = (int)fx0, y0 = (int)fy0;
  const int x1 = min(x0 + 1, 1023), y1 = min(y0 + 1, 1023);
  const float wx1 = 1.f - wx, wy1 = 1.f - wy;
  const float w00 = wx1 * wy1, w01 = wx * wy1;
  const float w10 = wx1 * wy,  w11 = wx * wy;
  const int i00 = (y0 << 10) + x0, i01 = (y0 << 10) + x1;
  const int i10 = (y1 << 10) + x0, i11 = (y1 << 10) + x1;
  const int obase = b * (16 * HW_) + hw;
#pragma unroll
  for (int c = 0; c < 16; ++c) {
    const float* pc = data + (size_t)c * TT_;
    const float t00 = fminf(fmaxf(pc[i00], CLAMP_LO), CLAMP_HI);
    const float t01 = fminf(fmaxf(pc[i01], CLAMP_LO), CLAMP_HI);
    const float t10 = fminf(fmaxf(pc[i10], CLAMP_LO), CLAMP_HI);
    const float t11 = fminf(fmaxf(pc[i11], CLAMP_LO), CLAMP_HI);
    const float r = fmaf(w11, t11, fmaf(w10, t10, fmaf(w01, t01, w00 * t00)));
    __builtin_nontemporal_store(r, out + obase + c * HW_);
  }
}

extern "C" void kernel_launch(void* const* d_in, const int* in_sizes, int n_in,
                              void* d_out, int out_size, void* d_ws,
                              size_t ws_size, hipStream_t stream) {
  (void)in_sizes; (void)n_in; (void)out_size;
  const float2* grid = (const float2*)d_in[0];   // x: [B,H,W,2]
  const float*  data = (const float*)d_in[1];    // data: [C,T,T]
  float* out = (float*)d_out;

  const size_t need = (size_t)TT_ * C_ * sizeof(float);   // 64 MB
  if (ws_size >= need) {
    tex_transpose_clamp<<<TT_ / 256, 256, 0, stream>>>(data, (float4*)d_ws);
    grid_sample_tex16<<<NPIX_ / 256, 256, 0, stream>>>(
        grid, (const float4*)d_ws, out);
  } else {
    grid_sample_direct<<<NPIX_ / 256, 256, 0, stream>>>(grid, data, out);
  }
}